// BPCALayer_74053826118348
// MI455X (gfx1250) — compile-verified
//
#include <hip/hip_runtime.h>

// ---------------------------------------------------------------------------
// BPCA pooling for x:[16,112,112,256] f32 -> out:[16,56,56,256] f32.
// "data" rows are 4 consecutive channels (input viewed as [B, HWC/4, 4]).
// Pass1: per-batch column stats (4 sums + 10 second moments), deterministic,
//        RT-cached streaming (we want these lines resident in the 192MB L2).
// Pass2: 4x4 Gram -> Jacobi eigensolve -> affine projection params a[4],bias.
// Pass3: projection via V_WMMA_F32_16X16X4_F32, walked in DESCENDING address
//        order so the second read of the input hits the L2 lines pass1 just
//        filled (LRU-friendly); NT loads (last use) + NT stores (write-once).
// ---------------------------------------------------------------------------

typedef __attribute__((ext_vector_type(2))) float v2f;
typedef __attribute__((ext_vector_type(4))) float v4f;
typedef __attribute__((ext_vector_type(8))) float v8f;

#define BATCH 16
#define HH 112
#define WW 112
#define CC 256
#define ROWS_PER_BATCH ((HH * WW * CC) / 4) /* 802816 float4 rows per batch */
#define BLOCKS_PER_BATCH 64
#define THREADS_P1 256
#define CHUNK (ROWS_PER_BATCH / BLOCKS_PER_BATCH) /* 12544 */
#define NSTAT 14
#define PARAM_OFF (BATCH * BLOCKS_PER_BATCH * NSTAT)
#define PARAM_STRIDE 8
#define TILES (BATCH * HH * WW * 4) /* one tile = 16 rows (16 c4 of one pixel) */

// ---------------- Pass 1: per-block partial sums (deterministic) ------------
__global__ void bpca_stats_partial(const float* __restrict__ x,
                                   float* __restrict__ ws) {
  const int b = blockIdx.x / BLOCKS_PER_BATCH;
  const int blk = blockIdx.x % BLOCKS_PER_BATCH;
  const int tid = threadIdx.x;
  const v4f* xr = reinterpret_cast<const v4f*>(x) +
                  (size_t)b * ROWS_PER_BATCH + (size_t)blk * CHUNK;

  float acc[NSTAT];
#pragma unroll
  for (int i = 0; i < NSTAT; ++i) acc[i] = 0.f;

  for (int i = tid; i < CHUNK; i += THREADS_P1) {
    // Prefetch a few strides ahead (global_prefetch_b8); keep RT temporal
    // hint on the loads themselves so lines stay in L2 for pass 3.
    __builtin_prefetch(xr + i + 4 * THREADS_P1, 0, 3);
    v4f v = xr[i]; // global_load_b128, fully coalesced stream
    acc[0] += v.x;       acc[1] += v.y;       acc[2] += v.z;       acc[3] += v.w;
    acc[4] += v.x * v.x; acc[5] += v.x * v.y; acc[6] += v.x * v.z; acc[7] += v.x * v.w;
    acc[8] += v.y * v.y; acc[9] += v.y * v.z; acc[10] += v.y * v.w;
    acc[11] += v.z * v.z; acc[12] += v.z * v.w; acc[13] += v.w * v.w;
  }

  __shared__ float red[THREADS_P1];
  float* part = ws + (size_t)(b * BLOCKS_PER_BATCH + blk) * NSTAT;
#pragma unroll 1
  for (int s = 0; s < NSTAT; ++s) {
    red[tid] = acc[s];
    __syncthreads();
    for (int off = THREADS_P1 / 2; off > 0; off >>= 1) {
      if (tid < off) red[tid] += red[tid + off];
      __syncthreads();
    }
    if (tid == 0) part[s] = red[0];
    __syncthreads();
  }
}

// ---------------- Pass 2: stats -> Gram -> eigenvector -> affine params ----
__global__ void bpca_solve(float* __restrict__ ws) {
  const int b = threadIdx.x;
  if (b >= BATCH) return;

  float st[NSTAT];
#pragma unroll 1
  for (int s = 0; s < NSTAT; ++s) {
    float v = 0.f;
    for (int i = 0; i < BLOCKS_PER_BATCH; ++i)
      v += ws[(size_t)(b * BLOCKS_PER_BATCH + i) * NSTAT + s];
    st[s] = v;
  }

  const float N = (float)ROWS_PER_BATCH;
  float Q[4][4];
  Q[0][0] = st[4];  Q[0][1] = st[5];  Q[0][2] = st[6];  Q[0][3] = st[7];
  Q[1][1] = st[8];  Q[1][2] = st[9];  Q[1][3] = st[10];
  Q[2][2] = st[11]; Q[2][3] = st[12]; Q[3][3] = st[13];
  Q[1][0] = Q[0][1]; Q[2][0] = Q[0][2]; Q[3][0] = Q[0][3];
  Q[2][1] = Q[1][2]; Q[3][1] = Q[1][3]; Q[3][2] = Q[2][3];

  float mean[4], inv_std[4];
  for (int j = 0; j < 4; ++j) {
    mean[j] = st[j] / N;
    float var = Q[j][j] / N - mean[j] * mean[j];
    var = var > 0.f ? var : 0.f;
    float sd = sqrtf(var);
    inv_std[j] = sd > 0.f ? 1.f / sd : 0.f; // std==0 => dn column 0 (NaN->0 rule)
  }

  // Gram of normalized data + identity eigenvector accumulator.
  float A[4][4], V[4][4];
  for (int i = 0; i < 4; ++i)
    for (int j = 0; j < 4; ++j) {
      A[i][j] = (Q[i][j] - N * mean[i] * mean[j]) * inv_std[i] * inv_std[j];
      V[i][j] = (i == j) ? 1.f : 0.f;
    }

  // Cyclic Jacobi sweeps on 4x4 symmetric matrix.
  for (int sweep = 0; sweep < 12; ++sweep) {
    for (int p = 0; p < 3; ++p) {
      for (int q = p + 1; q < 4; ++q) {
        float apq = A[p][q];
        if (fabsf(apq) <= 1e-20f) continue;
        float theta = (A[q][q] - A[p][p]) / (2.f * apq);
        float t = (theta >= 0.f ? 1.f : -1.f) /
                  (fabsf(theta) + sqrtf(theta * theta + 1.f));
        float c = 1.f / sqrtf(t * t + 1.f);
        float s = t * c;
        for (int k = 0; k < 4; ++k) {
          float Apk = A[p][k], Aqk = A[q][k];
          A[p][k] = c * Apk - s * Aqk;
          A[q][k] = s * Apk + c * Aqk;
        }
        for (int k = 0; k < 4; ++k) {
          float Akp = A[k][p], Akq = A[k][q];
          A[k][p] = c * Akp - s * Akq;
          A[k][q] = s * Akp + c * Akq;
        }
        for (int k = 0; k < 4; ++k) {
          float Vkp = V[k][p], Vkq = V[k][q];
          V[k][p] = c * Vkp - s * Vkq;
          V[k][q] = s * Vkp + c * Vkq;
        }
      }
    }
  }

  int dmax = 0;
  float emax = A[0][0];
  for (int d = 1; d < 4; ++d)
    if (A[d][d] > emax) { emax = A[d][d]; dmax = d; }

  float* prm = ws + PARAM_OFF + b * PARAM_STRIDE;
  float bias = 0.f;
  for (int j = 0; j < 4; ++j) {
    float aj = V[j][dmax] * inv_std[j]; // comp_j / std_j
    prm[j] = aj;
    bias -= aj * mean[j];
  }
  prm[4] = bias;
}

// ---------------- Pass 3: projection via V_WMMA_F32_16X16X4_F32 ------------
// Tile = 16 consecutive data rows (c4 = 16g..16g+15 of pixel (b,h,w)).
// A layout (ISA 7.12.2, 32-bit 16x4): lane L<16 -> (M=L, K=0..1),
// lanes 16..31 -> (M=L-16, K=2..3)  => one b64 load per lane, 256B contiguous.
// B = a[4] broadcast to all 16 columns; C preloaded with bias.
// D: lane 0 holds p0..p7 in d[0..7], lane 16 holds p8..p15 (all cols equal).
// Tiles are walked DESCENDING so we re-read the input in reverse of pass1's
// fill order -> maximal hits in the 192MB L2 (input is 205MB).
__global__ void bpca_project(const float* __restrict__ x,
                             const float* __restrict__ ws,
                             float* __restrict__ out) {
  const int lane = threadIdx.x & 31;
  const int l15 = lane & 15;
  const int half = lane >> 4; // 0 -> K=0,1 ; 1 -> K=2,3
  const int wavesPerBlock = blockDim.x >> 5;
  const int gwave = blockIdx.x * wavesPerBlock + (threadIdx.x >> 5);
  const int nwaves = gridDim.x * wavesPerBlock;

  int bprev = -1;
  v2f bmat;
  bmat.x = 0.f;
  bmat.y = 0.f;
  float bias = 0.f;

  for (int t = TILES - 1 - gwave; t >= 0; t -= nwaves) {
    const int g = t & 3;
    const int pix = t >> 2;
    const int w = pix % WW;
    const int rem = pix / WW;
    const int h = rem % HH;
    const int b = rem / HH;

    // Per-batch projection params; reload only on (wave-uniform) batch change.
    if (b != bprev) {
      bprev = b;
      const float* prm = ws + PARAM_OFF + b * PARAM_STRIDE;
      bmat.x = prm[2 * half + 0];
      bmat.y = prm[2 * half + 1];
      bias = prm[4];
    }

    const size_t in_base =
        ((size_t)(b * HH + h) * WW + w) * CC + (size_t)(64 * g);
    const size_t out_base =
        ((size_t)(b * (HH / 2) + (h >> 1)) * (WW / 2) + (w >> 1)) * CC +
        (size_t)(((h & 1) * 2 + (w & 1)) * 64 + 16 * g);

    // A: row m = l15, K pair selected by half (coalesced b64 per lane).
    // Last use of this input line -> nontemporal (hits L2 if resident,
    // then becomes replace-first instead of evicting unread lines).
    const v2f amat = __builtin_nontemporal_load(
        reinterpret_cast<const v2f*>(x + in_base + 4 * l15 + 2 * half));

    v8f cmat;
#pragma unroll
    for (int i = 0; i < 8; ++i) cmat[i] = bias;

    // D = A x B + C  (EXEC all ones here)
    v8f d = __builtin_amdgcn_wmma_f32_16x16x4_f32(
        /*neg_a=*/false, amat, /*neg_b=*/false, bmat,
        /*c_mod=*/(short)0, cmat, /*reuse_a=*/false, /*reuse_b=*/false);

    // Lanes 0 and 16 hold the 16 results; store 64 contiguous bytes.
    // Output is write-once/never-re-read -> nontemporal stores.
    if (l15 == 0) {
      float* o = out + out_base + half * 8;
      v4f s0;
      s0.x = d[0]; s0.y = d[1]; s0.z = d[2]; s0.w = d[3];
      v4f s1;
      s1.x = d[4]; s1.y = d[5]; s1.z = d[6]; s1.w = d[7];
      __builtin_nontemporal_store(s0, reinterpret_cast<v4f*>(o));
      __builtin_nontemporal_store(s1, reinterpret_cast<v4f*>(o + 4));
    }
  }
}

// ---------------------------------------------------------------------------
extern "C" void kernel_launch(void* const* d_in, const int* in_sizes, int n_in,
                              void* d_out, int out_size, void* d_ws,
                              size_t ws_size, hipStream_t stream) {
  (void)in_sizes; (void)n_in; (void)out_size; (void)ws_size;
  const float* x = (const float*)d_in[0];
  float* out = (float*)d_out;
  float* ws = (float*)d_ws; // [16*64*14 partials][16*8 params] ~58KB, no init needed

  bpca_stats_partial<<<BATCH * BLOCKS_PER_BATCH, THREADS_P1, 0, stream>>>(x, ws);
  bpca_solve<<<1, 32, 0, stream>>>(ws);
  bpca_project<<<1024, 256, 0, stream>>>(x, ws, out);
}